// WDSAC_15822659519168
// MI455X (gfx1250) — compile-verified
//
#include <hip/hip_runtime.h>
#include <math.h>

#define B_ 64
#define N_ 1024
#define G_ 256
#define H_ 512
// 1/(2*SIGMA2) with SIGMA2 = 0.01
#define INV2SIG2 50.0f
#define ALPHA 0.5f

typedef __attribute__((ext_vector_type(2))) float v2f;
typedef __attribute__((ext_vector_type(8))) float v8f;

// ---------------------------------------------------------------------------
// Kernel 1: exact stable top-K (descending, ties -> lower index) + gather.
// rank_i = #{j: w_j > w_i} + #{j < i: w_j == w_i}; elements with rank < G are
// the top-G and land at output row `rank` (matches jax.lax.top_k ordering).
// ---------------------------------------------------------------------------
__global__ __launch_bounds__(1024) void topk_gather_kernel(
    const float* __restrict__ pts,   // (B,3,N)
    const float* __restrict__ w,     // (B,N)
    float* __restrict__ gpts)        // (B,G,3)
{
    __shared__ float ws[N_];
    const int b = blockIdx.x;
    const int i = threadIdx.x;
    ws[i] = w[b * N_ + i];
    __syncthreads();
    const float wi = ws[i];
    int rank = 0;
    #pragma unroll 4
    for (int j = 0; j < N_; ++j) {
        const float wj = ws[j];
        rank += (int)((wj > wi) || ((wj == wi) && (j < i)));
    }
    if (rank < G_) {
        float* o = gpts + ((size_t)b * G_ + rank) * 3;
        o[0] = pts[b * 3 * N_ + 0 * N_ + i];
        o[1] = pts[b * 3 * N_ + 1 * N_ + i];
        o[2] = pts[b * 3 * N_ + 2 * N_ + i];
    }
}

// ---------------------------------------------------------------------------
// Kernel 2: fused planes + WMMA scoring + argmax/softmax reductions.
// One block (16 wave32) per batch.
// ---------------------------------------------------------------------------
__global__ __launch_bounds__(512) void wdsac_main_kernel(
    const float* __restrict__ pts,     // (B,3,N)
    const float* __restrict__ target,  // (B,3)
    const int*   __restrict__ sel,     // (B,H,3)
    const float* __restrict__ gpts,    // (B,G,3)
    float* __restrict__ exp_loss,      // (B)
    float* __restrict__ top_loss,      // (B)
    float* __restrict__ pred)          // (B,3)
{
    __shared__ float ptsL[3][N_];      // 12 KB  points, SoA
    __shared__ float planesL[H_][4];   //  8 KB  [nx,ny,nz,d] normalized
    __shared__ float lossL[H_];        //  2 KB
    __shared__ float scoreL[H_];       //  2 KB
    __shared__ float redV[512];        //  2 KB
    __shared__ int   redI[512];        //  2 KB
    __shared__ float redW[512];        //  2 KB

    const int t = threadIdx.x;
    const int b = blockIdx.x;

    // ---- phase 0: stage this batch's points into LDS -----------------------
    for (int k = t; k < 3 * N_; k += 512)
        (&ptsL[0][0])[k] = pts[(size_t)b * 3 * N_ + k];

    // ---- phase 1: one hypothesis per thread -> plane + loss ----------------
    {
        const int h = t;
        const int* s = sel + ((size_t)b * H_ + h) * 3;
        const float* gb = gpts + (size_t)b * G_ * 3;
        const float* q0 = gb + s[0] * 3;
        const float* q1 = gb + s[1] * 3;
        const float* q2 = gb + s[2] * 3;
        const float e0x = q1[0] - q0[0], e0y = q1[1] - q0[1], e0z = q1[2] - q0[2];
        const float e1x = q2[0] - q0[0], e1y = q2[1] - q0[1], e1z = q2[2] - q0[2];
        float nx = e0y * e1z - e0z * e1y;
        float ny = e0z * e1x - e0x * e1z;
        float nz = e0x * e1y - e0y * e1x;
        float d  = -(nx * q0[0] + ny * q0[1] + nz * q0[2]);
        if (nx == 0.0f && ny == 0.0f && nz == 0.0f && d == 0.0f) {
            nx = 1.0f; ny = 1.0f; nz = 1.0f; d = 1.0f;
        }
        const float inv = 1.0f / sqrtf(nx * nx + ny * ny + nz * nz);
        nx *= inv; ny *= inv; nz *= inv; d *= inv;
        planesL[h][0] = nx; planesL[h][1] = ny;
        planesL[h][2] = nz; planesL[h][3] = d;
        const float tx = target[b * 3 + 0];
        const float ty = target[b * 3 + 1];
        const float tz = target[b * 3 + 2];
        const float lm = (nx - tx) * (nx - tx) + (ny - ty) * (ny - ty) + (nz - tz) * (nz - tz);
        const float lp = (nx + tx) * (nx + tx) + (ny + ty) * (ny + ty) + (nz + tz) * (nz + tz);
        lossL[h] = fminf(lm, lp);
    }
    __syncthreads();

    // ---- phase 2: WMMA scoring: D = [n|d](16x4) x [p;1](4x16) --------------
    // A layout (16x4 f32): lanes 0-15 -> {K0,K1}, lanes 16-31 -> {K2,K3}, M = lane&15
    // B layout (4x16 f32): lanes 0-15 -> {K0,K1}, lanes 16-31 -> {K2,K3}, N = lane&15
    // C/D layout: VGPR r -> M = r (lanes 0-15) / r+8 (lanes 16-31), N = lane&15
    const int wave = t >> 5;
    const int lane = t & 31;
    const bool hi  = lane >= 16;
    const int  col = lane & 15;
    const float* pb0 = hi ? &ptsL[2][col] : &ptsL[0][col];
    const float* pb1 = &ptsL[1][col];

    for (int tile = wave; tile < H_ / 16; tile += 16) {
        const int hb = tile * 16;
        const int h  = hb + col;
        v2f a;
        a[0] = hi ? planesL[h][2] : planesL[h][0];
        a[1] = hi ? planesL[h][3] : planesL[h][1];

        float acc[8] = {0.f, 0.f, 0.f, 0.f, 0.f, 0.f, 0.f, 0.f};
        for (int n0 = 0; n0 < N_; n0 += 16) {
            // Unconditional LDS loads for all lanes; select by VALU cndmask
            // (keeps EXEC all-ones through the loop -- no saveexec around the
            // ds_load, and WMMA requires EXEC==all-1s anyway).
            const float p0v = pb0[n0];
            const float p1v = pb1[n0];
            v2f bv;
            bv[0] = p0v;
            bv[1] = hi ? 1.0f : p1v;
            v8f c = {};
            c = __builtin_amdgcn_wmma_f32_16x16x4_f32(
                    /*neg_a=*/false, a, /*neg_b=*/false, bv,
                    /*c_mod=*/(short)0, c, /*reuse_a=*/false, /*reuse_b=*/false);
            #pragma unroll
            for (int r = 0; r < 8; ++r) {
                const float v = c[r];
                acc[r] += __expf(-INV2SIG2 * v * v);  // |v|^2 == v^2
            }
        }
        // reduce over the 16 lanes of each half (stays within half: masks<16)
        #pragma unroll
        for (int r = 0; r < 8; ++r) {
            acc[r] += __shfl_xor(acc[r], 1, 32);
            acc[r] += __shfl_xor(acc[r], 2, 32);
            acc[r] += __shfl_xor(acc[r], 4, 32);
            acc[r] += __shfl_xor(acc[r], 8, 32);
        }
        if (lane == 0) {
            #pragma unroll
            for (int r = 0; r < 8; ++r) scoreL[hb + r] = acc[r];
        }
        if (lane == 16) {
            #pragma unroll
            for (int r = 0; r < 8; ++r) scoreL[hb + 8 + r] = acc[r];
        }
    }
    __syncthreads();

    // ---- phase 3: argmax (first-max tie break) -----------------------------
    redV[t] = scoreL[t];
    redI[t] = t;
    __syncthreads();
    for (int s = 256; s > 0; s >>= 1) {
        if (t < s) {
            const float v2 = redV[t + s];
            const int   i2 = redI[t + s];
            if (v2 > redV[t] || (v2 == redV[t] && i2 < redI[t])) {
                redV[t] = v2; redI[t] = i2;
            }
        }
        __syncthreads();
    }
    const float smax = redV[0];
    const int   amax = redI[0];
    __syncthreads();

    // ---- softmax(ALPHA*score)-weighted loss --------------------------------
    const float e = __expf(ALPHA * (scoreL[t] - smax));
    redV[t] = e;
    redW[t] = e * lossL[t];
    __syncthreads();
    for (int s = 256; s > 0; s >>= 1) {
        if (t < s) { redV[t] += redV[t + s]; redW[t] += redW[t + s]; }
        __syncthreads();
    }
    if (t == 0) {
        exp_loss[b] = redW[0] / redV[0];
        top_loss[b] = lossL[amax];
        pred[b * 3 + 0] = planesL[amax][0];
        pred[b * 3 + 1] = planesL[amax][1];
        pred[b * 3 + 2] = planesL[amax][2];
    }
}

// ---------------------------------------------------------------------------
extern "C" void kernel_launch(void* const* d_in, const int* in_sizes, int n_in,
                              void* d_out, int out_size, void* d_ws, size_t ws_size,
                              hipStream_t stream) {
    const float* pts    = (const float*)d_in[0];  // (B,3,N)
    const float* target = (const float*)d_in[1];  // (B,3)
    const float* ptw    = (const float*)d_in[2];  // (B,N)
    const int*   sel    = (const int*)d_in[3];    // (B,H,3)

    float* out      = (float*)d_out;
    float* exp_loss = out;                 // 64
    float* top_loss = out + B_;            // 64
    float* pred     = out + 2 * B_;        // 192
    float* gpts     = out + 2 * B_ + 3 * B_;  // 64*256*3

    topk_gather_kernel<<<B_, N_, 0, stream>>>(pts, ptw, gpts);
    wdsac_main_kernel<<<B_, 512, 0, stream>>>(pts, target, sel, gpts,
                                              exp_loss, top_loss, pred);
}